// DynamicPartitionStitchModule_8057358648477
// MI455X (gfx1250) — compile-verified
//
#include <hip/hip_runtime.h>
#include <stdint.h>

// Problem constants (from the reference): N = 1<<20 rows, D = 64 f32 per row.
#define NTOT   (1 << 20)
#define DDIM   64
#define T      256                 // threads per block (8 wave32 waves)
#define CHUNK  2048                // rows per block
#define NB     (NTOT / CHUNK)      // 512 blocks
#define IPT    (CHUNK / T)         // 8 partition entries per thread
#define SUB    64                  // rows per copy subtile (64*256B = 16KB)
#define NSUB   (CHUNK / SUB)       // 32 subtiles
#define TASKS  (SUB * (DDIM / 4))  // 1024 float4 tasks per subtile
#define TPT    (TASKS / T)         // 4 tasks per thread

typedef float __attribute__((ext_vector_type(4))) v4f;
typedef int   __attribute__((ext_vector_type(4))) v4i;

// CDNA5 async global->LDS copy path (ASYNCcnt-tracked). Guarded so the file
// always compiles; fallback is a direct b128 register copy.
#if defined(__HIP_DEVICE_COMPILE__) && defined(__gfx1250__) && \
    __has_builtin(__builtin_amdgcn_global_load_async_to_lds_b128)
#define HAVE_ASYNC 1
#else
#define HAVE_ASYNC 0
#endif

#if HAVE_ASYNC
typedef __attribute__((address_space(1))) v4i global_v4i;  // global int4
typedef __attribute__((address_space(3))) v4i shared_v4i;  // LDS int4

#if __has_builtin(__builtin_amdgcn_s_wait_asynccnt)
#define WAIT_ASYNC(n) __builtin_amdgcn_s_wait_asynccnt(n)
#else
#define WAIT_ASYNC(n) asm volatile("s_wait_asynccnt %0" ::"i"(n) : "memory")
#endif
#endif

// ---------------------------------------------------------------------------
// Kernel 1: per-block count of partition==0 entries.
// ---------------------------------------------------------------------------
__global__ void __launch_bounds__(T) k_count(const int* __restrict__ part,
                                             unsigned* __restrict__ cnt) {
    __shared__ unsigned s[T];
    const int t = threadIdx.x;
    const int base = blockIdx.x * CHUNK + t * IPT;
    v4i p0 = *(const v4i*)(part + base);
    v4i p1 = *(const v4i*)(part + base + 4);
    unsigned c = (unsigned)((p0.x == 0) + (p0.y == 0) + (p0.z == 0) + (p0.w == 0) +
                            (p1.x == 0) + (p1.y == 0) + (p1.z == 0) + (p1.w == 0));
    s[t] = c;
    __syncthreads();
    for (int off = T / 2; off > 0; off >>= 1) {
        if (t < off) s[t] += s[t + off];
        __syncthreads();
    }
    if (t == 0) cnt[blockIdx.x] = s[0];
}

// ---------------------------------------------------------------------------
// Kernel 2: exclusive scan of the NB block counts (single block, NB threads).
// ---------------------------------------------------------------------------
__global__ void __launch_bounds__(NB) k_scan(const unsigned* __restrict__ in,
                                             unsigned* __restrict__ out) {
    __shared__ unsigned s[NB];
    const int t = threadIdx.x;
    s[t] = in[t];
    __syncthreads();
    for (int off = 1; off < NB; off <<= 1) {
        unsigned v = (t >= off) ? s[t - off] : 0u;
        __syncthreads();
        s[t] += v;
        __syncthreads();
    }
    out[t] = t ? s[t - 1] : 0u;
}

// ---------------------------------------------------------------------------
// Kernel 3: compute destination row per source row, then scatter-copy rows.
// ---------------------------------------------------------------------------
__global__ void __launch_bounds__(T) k_stitch(const float* __restrict__ data,
                                              const int* __restrict__ part,
                                              const int* __restrict__ idx0,
                                              const int* __restrict__ idx1,
                                              const unsigned* __restrict__ blockZeros,
                                              float* __restrict__ out) {
    __shared__ int dstRow[CHUNK];     // 8 KB
    __shared__ unsigned tsum[T];      // 1 KB
#if HAVE_ASYNC
    __shared__ v4f buf[2][TASKS];     // 32 KB double-buffered stage
#endif
    const int t = threadIdx.x;
    const int blockBase = blockIdx.x * CHUNK;
    const int base = blockBase + t * IPT;

    // --- stable ranks within each partition ---
    v4i p0 = *(const v4i*)(part + base);
    v4i p1 = *(const v4i*)(part + base + 4);
    int pv[IPT] = {p0.x, p0.y, p0.z, p0.w, p1.x, p1.y, p1.z, p1.w};
    unsigned c = 0;
#pragma unroll
    for (int k = 0; k < IPT; ++k) c += (pv[k] == 0);

    tsum[t] = c;
    __syncthreads();
    for (int off = 1; off < T; off <<= 1) {       // inclusive scan of counts
        unsigned v = (t >= off) ? tsum[t - off] : 0u;
        __syncthreads();
        tsum[t] += v;
        __syncthreads();
    }
    // zeros strictly before this thread's first element:
    unsigned zb = blockZeros[blockIdx.x] + (tsum[t] - c);

#pragma unroll
    for (int k = 0; k < IPT; ++k) {
        const int i = base + k;                   // global source row
        int d;
        if (pv[k] == 0) { d = idx0[zb]; ++zb; }   // rank in partition 0 = zb
        else            { d = idx1[(unsigned)i - zb]; } // ones before i = i - zb
        dstRow[t * IPT + k] = d;
    }
    __syncthreads();

    // --- scatter-copy: 16 lanes x b128 per 256-byte row ---
#if HAVE_ASYNC
    auto issue = [&](int sub) {
        const int rowBase = blockBase + sub * SUB;
        v4f* bb = buf[sub & 1];
#pragma unroll
        for (int j = 0; j < TPT; ++j) {
            const int task = j * T + t;
            const int r = task >> 4, col = task & 15;
            const float* g = data + (size_t)(rowBase + r) * DDIM + col * 4;
            __builtin_amdgcn_global_load_async_to_lds_b128(
                (global_v4i*)(uintptr_t)g,
                (shared_v4i*)(uint32_t)(uintptr_t)&bb[task],
                0, 0);
        }
    };
    issue(0);
    for (int s = 0; s < NSUB; ++s) {
        if (s + 1 < NSUB) { issue(s + 1); WAIT_ASYNC(TPT); }  // keep next tile in flight
        else              { WAIT_ASYNC(0); }
        const v4f* bb = buf[s & 1];
#pragma unroll
        for (int j = 0; j < TPT; ++j) {
            const int task = j * T + t;
            const int r = task >> 4, col = task & 15;
            v4f v = bb[task];                      // thread reads only its own region
            const int dr = dstRow[s * SUB + r];
            __builtin_nontemporal_store(v, (v4f*)(out + (size_t)dr * DDIM + col * 4));
        }
    }
#else
    for (int s = 0; s < NSUB; ++s) {
        const int rowBase = blockBase + s * SUB;
#pragma unroll
        for (int j = 0; j < TPT; ++j) {
            const int task = j * T + t;
            const int r = task >> 4, col = task & 15;
            v4f v = __builtin_nontemporal_load(
                (const v4f*)(data + (size_t)(rowBase + r) * DDIM + col * 4));
            const int dr = dstRow[s * SUB + r];
            __builtin_nontemporal_store(v, (v4f*)(out + (size_t)dr * DDIM + col * 4));
        }
    }
#endif
}

// ---------------------------------------------------------------------------
extern "C" void kernel_launch(void* const* d_in, const int* in_sizes, int n_in,
                              void* d_out, int out_size, void* d_ws, size_t ws_size,
                              hipStream_t stream) {
    (void)in_sizes; (void)n_in; (void)out_size; (void)ws_size;
    const float* data = (const float*)d_in[0];
    const int*   part = (const int*)d_in[1];
    const int*   idx0 = (const int*)d_in[2];
    const int*   idx1 = (const int*)d_in[3];
    float*       out  = (float*)d_out;

    unsigned* cnt  = (unsigned*)d_ws;   // NB counts
    unsigned* excl = cnt + NB;          // NB exclusive prefixes (4 KB total ws)

    k_count <<<NB, T, 0, stream>>>(part, cnt);
    k_scan  <<<1, NB, 0, stream>>>(cnt, excl);
    k_stitch<<<NB, T, 0, stream>>>(data, part, idx0, idx1, excl, out);
}